// VQVAE_14912126451772
// MI455X (gfx1250) — compile-verified
//
#include <hip/hip_runtime.h>
#include <hip/hip_bf16.h>
#include <math.h>

typedef __attribute__((ext_vector_type(2))) float v2f;
typedef __attribute__((ext_vector_type(8))) float v8f;
typedef __attribute__((ext_vector_type(4))) unsigned int v4u;
typedef __attribute__((ext_vector_type(4))) int v4i;
typedef __attribute__((ext_vector_type(8))) int v8i;

#define NB   128
#define NT   128
#define NTOK (NB*NT)          // 16384
#define HID  128
#define EMB  64
#define KCB  8192
#define VSP  2000
#define VLC  32
#define CE   6
#define FUSE_IN 384

// ---- workspace offsets (bytes) ----
#define WS_NAIP  0            // 128*128*4        = 65536
#define WS_ZE    65536        // 16384*64*4       = 4194304
#define WS_ESQ   4259840      // 8192*4           = 32768
#define WS_IDX   4292608      // 16384*4          = 65536
#define WS_CNT   4358144      // 8192*4           = 32768
#define WS_DTOK  4390912      // 16384*4          = 65536
#define WS_H     4456448      // 16384*128*4      = 8388608  (end ~12.3 MB)

// ---- output offsets (floats, reference tuple order) ----
#define OFF_CONT 0            // (B,T,16)   262144
#define OFF_SP   262144       // (B,T,2000) 32768000
#define OFF_LC   33030144     // (B,T,32)   524288
#define OFF_CAN  33554432     // (B,)       128
#define OFF_IDX  33554560     // (B,T)      16384
#define OFF_LOSS 33570944
#define OFF_PERP 33570945

static __device__ __forceinline__ float f0(float v) {
  return __builtin_isfinite(v) ? v : 0.0f;
}

// exact fp32 matrix op: D(16x16) += A(16x4) x B(4x16)
static __device__ __forceinline__ v8f wmma_f32_k4(v2f a, v2f b, v8f c) {
  return __builtin_amdgcn_wmma_f32_16x16x4_f32(false, a, false, b, (short)0, c,
                                               false, false);
}

// ---------------- TDM: 2D fp32 tile (global -> LDS) ----------------
// Builds a Tensor DMA Descriptor (D#) per CDNA5 ISA and
// issues TENSOR_LOAD_TO_LDS. data_size = 4 bytes (code 2). 2-D tensor:
// groups 2/3 zero. lds_off is a byte offset into the LDS segment (we use
// dynamic LDS only in the calling kernels, so the tile base offset is known).
static __device__ __forceinline__ void tdm_load_2d_f32(
    unsigned lds_off, const float* gptr,
    unsigned tensor_w, unsigned tensor_h,
    unsigned tile_w, unsigned tile_h) {
  unsigned long long ga = (unsigned long long)(const void*)gptr;
  // group0: [1:0]=count=1 | [63:32]=lds_addr | [120:64]=global_addr | [127:126]=type 2
  v4u g0 = { 1u,
             lds_off,
             (unsigned)(ga & 0xffffffffu),
             0x80000000u | (unsigned)((ga >> 32) & 0x1ffffffu) };
  // group1: [15:0]=wg_mask 0 | [17:16]=data_size 2 | [79:48]=tensor_dim0 |
  //         [111:80]=tensor_dim1 | [127:112]=tile_dim0 | [143:128]=tile_dim1 |
  //         [207:160]=tensor_dim0_stride (= tensor_w, 48-bit)
  v8i g1 = { (int)0x20000u,
             (int)(tensor_w << 16),
             (int)((tensor_w >> 16) | (tensor_h << 16)),
             (int)((tensor_h >> 16) | (tile_w << 16)),
             (int)(tile_h & 0xffffu),
             (int)tensor_w,      // dim0 stride, low 32
             0, 0 };
  v4i gz = { 0, 0, 0, 0 };
#if __clang_major__ >= 23
  v8i gz8 = { 0, 0, 0, 0, 0, 0, 0, 0 };
  __builtin_amdgcn_tensor_load_to_lds(g0, g1, gz, gz, gz8, 0);
#else
  __builtin_amdgcn_tensor_load_to_lds(g0, g1, gz, gz, 0);
#endif
}

// ---------------- init: zero codebook usage counts ----------------
__global__ void init_kernel(int* __restrict__ counts) {
  int i = blockIdx.x * blockDim.x + threadIdx.x;
  if (i < KCB) counts[i] = 0;
}

// ---------------- ||e_j||^2 per code ----------------
__global__ void esq_kernel(const float* __restrict__ cb, float* __restrict__ e_sq) {
  int i = blockIdx.x * blockDim.x + threadIdx.x;
  if (i < KCB) {
    const float* r = cb + (size_t)i * EMB;
    float s = 0.f;
    #pragma unroll
    for (int j = 0; j < EMB; ++j) s += r[j] * r[j];
    e_sq[i] = s;
  }
}

// ---------------- naip: 3x3 conv (c=1) + relu + proj ----------------
__global__ void naip_kernel(const float* __restrict__ naip,
                            const float* __restrict__ cw, const float* __restrict__ cbv,
                            const float* __restrict__ pw, const float* __restrict__ pb,
                            float* __restrict__ naip_feat) {
  __shared__ float s_n[9];
  __shared__ float s_f[HID];
  int b = blockIdx.x, tid = threadIdx.x;
  if (tid < 9) s_n[tid] = f0(naip[b * 9 + tid]);
  __syncthreads();
  float acc = cbv[tid];
  const float* wr = cw + tid * 9;
  #pragma unroll
  for (int j = 0; j < 9; ++j) acc += wr[j] * s_n[j];
  s_f[tid] = fmaxf(acc, 0.f);
  __syncthreads();
  float o = pb[tid];
  const float* pr = pw + tid * HID;
  for (int j = 0; j < HID; ++j) o += pr[j] * s_f[j];
  naip_feat[(size_t)b * HID + tid] = o;
}

// ---------------- per-token encoder -> z_e ----------------
__global__ void enc_kernel(const float* __restrict__ cont, const int* __restrict__ cat,
                           const float* __restrict__ emb_sp, const float* __restrict__ emb_lc,
                           const float* __restrict__ cpw, const float* __restrict__ cpb,
                           const float* __restrict__ kpw, const float* __restrict__ kpb,
                           const float* __restrict__ f1w, const float* __restrict__ f1b,
                           const float* __restrict__ f2w, const float* __restrict__ f2b,
                           const float* __restrict__ naip_feat, float* __restrict__ z_e) {
  __shared__ float s_in[28];       // 16 cont + 12 cat-embed
  __shared__ float s_f[FUSE_IN];
  __shared__ float s_h1[HID];
  int tok = blockIdx.x, tid = threadIdx.x;
  int b = tok >> 7;
  if (tid < 16) s_in[tid] = f0(cont[(size_t)tok * 16 + tid]);
  if (tid < 12) {
    float v;
    if (tid < 6) { int i0 = cat[(size_t)tok * 2 + 0]; v = emb_sp[(size_t)i0 * CE + tid]; }
    else         { int i1 = cat[(size_t)tok * 2 + 1]; v = emb_lc[(size_t)i1 * CE + (tid - 6)]; }
    s_in[16 + tid] = v;
  }
  __syncthreads();
  {
    float ch = cpb[tid];
    const float* w = cpw + tid * 16;
    #pragma unroll
    for (int j = 0; j < 16; ++j) ch += w[j] * s_in[j];
    float ca = kpb[tid];
    const float* w2 = kpw + tid * 12;
    #pragma unroll
    for (int j = 0; j < 12; ++j) ca += w2[j] * s_in[16 + j];
    s_f[tid]       = naip_feat[(size_t)b * HID + tid];
    s_f[HID + tid] = ch;
    s_f[2 * HID + tid] = ca;
  }
  __syncthreads();
  {
    float r = f1b[tid];
    const float* w = f1w + (size_t)tid * FUSE_IN;
    for (int j = 0; j < FUSE_IN; ++j) r += w[j] * s_f[j];
    s_h1[tid] = fmaxf(r, 0.f);
  }
  __syncthreads();
  if (tid < EMB) {
    float z = f2b[tid];
    const float* w = f2w + tid * HID;
    for (int j = 0; j < HID; ++j) z += w[j] * s_h1[j];
    z_e[(size_t)tok * EMB + tid] = z;
  }
}

// ---------------- fused VQ distance GEMM + argmin (fp32 WMMA + TDM) ----------
// One block handles 16 token rows; 8 waves stripe over the 512 codebook col-tiles.
// z tile staged into LDS by the Tensor Data Mover; argmin uses
// score = ||e||^2 - 2*z.e (row-constant ||z||^2 dropped).
// Dynamic LDS layout: [0,4096) z tile | [4096,4608) minval | [4608,5120) minidx
__global__ void __launch_bounds__(256) vq_kernel(const float* __restrict__ z_e,
                                                 const float* __restrict__ cb,
                                                 const float* __restrict__ e_sq,
                                                 int* __restrict__ idx_out) {
  extern __shared__ char smem[];
  float* s_z    = (float*)smem;
  float* s_min  = (float*)(smem + 4096);
  int*   s_mini = (int*)  (smem + 4608);
  int tid = threadIdx.x;
  int row_base = blockIdx.x * 16;

  if (tid < 32) {   // wave 0 issues the TDM copy (EXEC-independent, wave-scoped)
    tdm_load_2d_f32(/*lds_off=*/0u, z_e + (size_t)row_base * EMB,
                    /*tensor_w=*/EMB, /*tensor_h=*/16,
                    /*tile_w=*/EMB, /*tile_h=*/16);
    __builtin_amdgcn_s_wait_tensorcnt(0);
  }
  __syncthreads();

  int wave = tid >> 5, lane = tid & 31, half = lane >> 4, ln = lane & 15;
  float mval[8]; int midx[8];
  #pragma unroll
  for (int v = 0; v < 8; ++v) { mval[v] = INFINITY; midx[v] = 0; }

  const float* aptr = s_z + ln * EMB + half * 2;   // A[M=ln][k + half*2 ..]
  for (int ct = wave; ct < KCB / 16; ct += 8) {    // wave-uniform -> EXEC all-1s
    int col = ct * 16 + ln;                        // this lane's N column
    const float* bptr = cb + (size_t)col * EMB + half * 2;
    // prefetch next tile's B column for this lane (speculative, L2)
    int pc = (ct + 8 < KCB / 16) ? ct + 8 : ct;
    __builtin_prefetch(cb + (size_t)(pc * 16 + ln) * EMB, 0, 3);
    v8f c = {0.f, 0.f, 0.f, 0.f, 0.f, 0.f, 0.f, 0.f};
    #pragma unroll
    for (int kk = 0; kk < EMB / 4; ++kk) {         // 16 x K4 steps
      v2f a = *(const v2f*)(aptr + kk * 4);
      v2f b = *(const v2f*)(bptr + kk * 4);
      c = wmma_f32_k4(a, b, c);
    }
    float es = e_sq[col];
    #pragma unroll
    for (int v = 0; v < 8; ++v) {
      float score = es - 2.0f * c[v];
      if (score < mval[v]) { mval[v] = score; midx[v] = col; }
    }
  }
  // reduce across the 16 lanes (same M, different N) in each half-wave
  #pragma unroll
  for (int v = 0; v < 8; ++v) {
    float mv = mval[v]; int mi = midx[v];
    #pragma unroll
    for (int m = 1; m < 16; m <<= 1) {
      float ov = __shfl_xor(mv, m, 16);
      int   oi = __shfl_xor(mi, m, 16);
      if (ov < mv || (ov == mv && oi < mi)) { mv = ov; mi = oi; }
    }
    if (ln == 0) {
      int row = v + half * 8;                      // half 0 -> M=v, half 1 -> M=v+8
      s_min[row * 8 + wave] = mv;
      s_mini[row * 8 + wave] = mi;
    }
  }
  __syncthreads();
  if (tid < 16) {
    float bv = s_min[tid * 8]; int bi = s_mini[tid * 8];
    #pragma unroll
    for (int w = 1; w < 8; ++w) {
      float v = s_min[tid * 8 + w]; int i = s_mini[tid * 8 + w];
      if (v < bv || (v == bv && i < bi)) { bv = v; bi = i; }
    }
    idx_out[row_base + tid] = bi;
  }
}

// ---------------- decoder: gather z_q, bb1/bb2, small heads, loss terms ------
__global__ void dec_kernel(const float* __restrict__ z_e, const float* __restrict__ cb,
                           const int* __restrict__ idx,
                           const float* __restrict__ b1w, const float* __restrict__ b1b,
                           const float* __restrict__ b2w, const float* __restrict__ b2b,
                           const float* __restrict__ chw, const float* __restrict__ chb,
                           const float* __restrict__ lcw, const float* __restrict__ lcb,
                           float* __restrict__ h_ws, int* __restrict__ counts,
                           float* __restrict__ d_tok, float* __restrict__ out) {
  __shared__ float s_zq[EMB];
  __shared__ float s_t1[HID];
  __shared__ float s_h[HID];
  __shared__ float s_red[EMB];
  int tok = blockIdx.x, tid = threadIdx.x;
  int ci = idx[tok];
  if (tid < EMB) s_zq[tid] = cb[(size_t)ci * EMB + tid];
  __syncthreads();
  {
    float a = b1b[tid];
    const float* w = b1w + tid * EMB;
    #pragma unroll
    for (int j = 0; j < EMB; ++j) a += w[j] * s_zq[j];
    s_t1[tid] = fmaxf(a, 0.f);
  }
  if (tid < EMB) {
    float d = z_e[(size_t)tok * EMB + tid] - s_zq[tid];
    s_red[tid] = d * d;
  }
  __syncthreads();
  {
    float a = b2b[tid];
    const float* w = b2w + tid * HID;
    for (int j = 0; j < HID; ++j) a += w[j] * s_t1[j];
    float hv = fmaxf(a, 0.f);
    s_h[tid] = hv;
    h_ws[(size_t)tok * HID + tid] = hv;
  }
  __syncthreads();
  if (tid == 0) {
    float s = 0.f;
    #pragma unroll
    for (int j = 0; j < EMB; ++j) s += s_red[j];
    d_tok[tok] = s;                                // ||z_e - z_q||^2
    atomicAdd(&counts[ci], 1);
    out[OFF_IDX + tok] = (float)ci;
  }
  if (tid < 16) {
    float a = chb[tid];
    const float* w = chw + tid * HID;
    for (int j = 0; j < HID; ++j) a += w[j] * s_h[j];
    out[OFF_CONT + (size_t)tok * 16 + tid] = a;
  }
  if (tid < 32) {
    float a = lcb[tid];
    const float* w = lcw + tid * HID;
    for (int j = 0; j < HID; ++j) a += w[j] * s_h[j];
    out[OFF_LC + (size_t)tok * 32 + tid] = a;
  }
}

// ---------------- sp_logits: (16384x128) x (128x2000) fp32 WMMA + TDM --------
// blockIdx.y picks 16 token rows (TDM-staged into LDS); 8 waves each own one
// 16-col tile (125 total). Output-store bandwidth bound (131 MB).
__global__ void __launch_bounds__(256) sp_kernel(const float* __restrict__ h,
                                                 const float* __restrict__ w,
                                                 const float* __restrict__ bias,
                                                 float* __restrict__ out) {
  extern __shared__ char smem[];
  float* s_h = (float*)smem;                       // 16*128 fp32 at LDS offset 0
  int tid = threadIdx.x;
  int row_base = blockIdx.y * 16;

  if (tid < 32) {
    tdm_load_2d_f32(/*lds_off=*/0u, h + (size_t)row_base * HID,
                    /*tensor_w=*/HID, /*tensor_h=*/16,
                    /*tile_w=*/HID, /*tile_h=*/16);
    __builtin_amdgcn_s_wait_tensorcnt(0);
  }
  __syncthreads();

  int wave = tid >> 5, lane = tid & 31, half = lane >> 4, ln = lane & 15;
  int ntile = blockIdx.x * 8 + wave;
  if (ntile >= VSP / 16) return;                   // wave-uniform exit, EXEC stays full
  int col = ntile * 16 + ln;

  const float* aptr = s_h + ln * HID + half * 2;
  const float* bptr = w + (size_t)col * HID + half * 2;
  v8f c = {0.f, 0.f, 0.f, 0.f, 0.f, 0.f, 0.f, 0.f};
  #pragma unroll
  for (int kk = 0; kk < HID / 4; ++kk) {           // 32 x K4 steps
    v2f a = *(const v2f*)(aptr + kk * 4);
    v2f b = *(const v2f*)(bptr + kk * 4);
    c = wmma_f32_k4(a, b, c);
  }
  float bv = bias[col];
  #pragma unroll
  for (int v = 0; v < 8; ++v) {
    int row = row_base + v + half * 8;
    out[(size_t)row * VSP + col] = c[v] + bv;
  }
}

// ---------------- canopy head on h[:, T-1, :] ----------------
__global__ void can_kernel(const float* __restrict__ h_ws,
                           const float* __restrict__ c1w, const float* __restrict__ c1b,
                           const float* __restrict__ c2w, const float* __restrict__ c2b,
                           float* __restrict__ out) {
  __shared__ float s_h[HID];
  __shared__ float s_t[HID];
  int b = blockIdx.x, tid = threadIdx.x;
  s_h[tid] = h_ws[((size_t)b * NT + (NT - 1)) * HID + tid];
  __syncthreads();
  float a = c1b[tid];
  const float* w = c1w + tid * HID;
  for (int j = 0; j < HID; ++j) a += w[j] * s_h[j];
  s_t[tid] = fmaxf(a, 0.f);
  __syncthreads();
  if (tid == 0) {
    float acc = c2b[0];
    for (int j = 0; j < HID; ++j) acc += c2w[j] * s_t[j];
    out[OFF_CAN + b] = acc;
  }
}

// ---------------- finalize: loss_vq + perplexity (deterministic) ----------------
__global__ void fin_kernel(const float* __restrict__ d_tok, const int* __restrict__ counts,
                           float* __restrict__ out) {
  __shared__ float sd[256];
  __shared__ float sp[256];
  int tid = threadIdx.x;
  float a = 0.f;
  for (int i = tid; i < NTOK; i += 256) a += d_tok[i];
  float b = 0.f;
  for (int i = tid; i < KCB; i += 256) {
    float p = (float)counts[i] / (float)NTOK;
    b += p * logf(p + 1e-12f);
  }
  sd[tid] = a; sp[tid] = b;
  __syncthreads();
  for (int s = 128; s > 0; s >>= 1) {
    if (tid < s) { sd[tid] += sd[tid + s]; sp[tid] += sp[tid + s]; }
    __syncthreads();
  }
  if (tid == 0) {
    // forward: commitment == codebook_l == MSE -> loss = (1 + BETA) * MSE
    out[OFF_LOSS] = 1.25f * sd[0] / (float)(NTOK * EMB);
    out[OFF_PERP] = expf(-sp[0]);
  }
}

extern "C" void kernel_launch(void* const* d_in, const int* in_sizes, int n_in,
                              void* d_out, int out_size, void* d_ws, size_t ws_size,
                              hipStream_t stream) {
  (void)in_sizes; (void)n_in; (void)out_size; (void)ws_size;
  const float* cont   = (const float*)d_in[0];
  const float* naip   = (const float*)d_in[1];
  const int*   cat    = (const int*)  d_in[2];
  const float* emb_sp = (const float*)d_in[3];
  const float* emb_lc = (const float*)d_in[4];
  const float* conv_w = (const float*)d_in[5];
  const float* conv_b = (const float*)d_in[6];
  const float* npw    = (const float*)d_in[7];
  const float* npb    = (const float*)d_in[8];
  const float* cpw    = (const float*)d_in[9];
  const float* cpb    = (const float*)d_in[10];
  const float* kpw    = (const float*)d_in[11];
  const float* kpb    = (const float*)d_in[12];
  const float* f1w    = (const float*)d_in[13];
  const float* f1b    = (const float*)d_in[14];
  const float* f2w    = (const float*)d_in[15];
  const float* f2b    = (const float*)d_in[16];
  const float* cb     = (const float*)d_in[17];
  const float* b1w    = (const float*)d_in[18];
  const float* b1b    = (const float*)d_in[19];
  const float* b2w    = (const float*)d_in[20];
  const float* b2b    = (const float*)d_in[21];
  const float* chw    = (const float*)d_in[22];
  const float* chb    = (const float*)d_in[23];
  const float* spw    = (const float*)d_in[24];
  const float* spb    = (const float*)d_in[25];
  const float* lcw    = (const float*)d_in[26];
  const float* lcb    = (const float*)d_in[27];
  const float* c1w    = (const float*)d_in[28];
  const float* c1b    = (const float*)d_in[29];
  const float* c2w    = (const float*)d_in[30];
  const float* c2b    = (const float*)d_in[31];

  char* ws = (char*)d_ws;
  float* naip_feat = (float*)(ws + WS_NAIP);
  float* z_e       = (float*)(ws + WS_ZE);
  float* e_sq      = (float*)(ws + WS_ESQ);
  int*   idx       = (int*)  (ws + WS_IDX);
  int*   counts    = (int*)  (ws + WS_CNT);
  float* d_tok     = (float*)(ws + WS_DTOK);
  float* h_ws      = (float*)(ws + WS_H);
  float* out       = (float*)d_out;

  init_kernel<<<KCB / 256, 256, 0, stream>>>(counts);
  esq_kernel <<<KCB / 256, 256, 0, stream>>>(cb, e_sq);
  naip_kernel<<<NB, HID, 0, stream>>>(naip, conv_w, conv_b, npw, npb, naip_feat);
  enc_kernel <<<NTOK, HID, 0, stream>>>(cont, cat, emb_sp, emb_lc, cpw, cpb,
                                        kpw, kpb, f1w, f1b, f2w, f2b,
                                        naip_feat, z_e);
  vq_kernel  <<<NTOK / 16, 256, 5120, stream>>>(z_e, cb, e_sq, idx);
  dec_kernel <<<NTOK, HID, 0, stream>>>(z_e, cb, idx, b1w, b1b, b2w, b2b,
                                        chw, chb, lcw, lcb, h_ws, counts,
                                        d_tok, out);
  sp_kernel  <<<dim3(16, NTOK / 16), 256, 8192, stream>>>(h_ws, spw, spb, out + OFF_SP);
  can_kernel <<<NB, HID, 0, stream>>>(h_ws, c1w, c1b, c2w, c2b, out);
  fin_kernel <<<1, 256, 0, stream>>>(d_tok, counts, out);
}